// glimpse_network_46634754900350
// MI455X (gfx1250) — compile-verified
//
#include <hip/hip_runtime.h>
#include <hip/hip_bf16.h>

// ---------------------------------------------------------------------------
// Glimpse network for MI455X (gfx1250, wave32).
//  Phase 1: foveate (memory-bound, early-exit scan)        -> phi  (B,300)
//  Phase 2: relu(phi@W1^T+b1) via V_WMMA_F32_16X16X4_F32   -> A2[:, :128]
//  Phase 3: relu(l@W2^T+b2)   (K=3, scalar)                -> A2[:,128:256]
//  Phase 4: relu(A2@[W3|W4]^T + b3+b4) via WMMA f32        -> out (B,256)
// f32 WMMA keeps full precision vs. the f32 reference while still using the
// CDNA5 matrix pipe.
// ---------------------------------------------------------------------------

#define PP 10000
#define NPTS 100
#define BHALF 0.25f

typedef float v2f __attribute__((ext_vector_type(2)));
typedef float v8f __attribute__((ext_vector_type(8)));

__device__ __forceinline__ v8f wmma_f32_k4(v2f a, v2f b, v8f c) {
    // 8 args: (neg_a, A, neg_b, B, c_mod, C, reuse_a, reuse_b)
    return __builtin_amdgcn_wmma_f32_16x16x4_f32(
        false, a, false, b, (short)0, c, false, false);
}

// ---------------- Phase 1: foveate ----------------
// One 256-thread block (8 waves) per cloud. Stable rank via ballot+prefix.
__global__ void foveate_kernel(const float* __restrict__ x,
                               const float* __restrict__ l,
                               float* __restrict__ phi)
{
    const int b    = blockIdx.x;
    const int tid  = threadIdx.x;
    const int lane = tid & 31;
    const int wave = tid >> 5;

    __shared__ float s_sel[3 * NPTS];
    __shared__ int   s_wcnt[8];

    const float lx = l[b * 3 + 0], ly = l[b * 3 + 1], lz = l[b * 3 + 2];
    const float lox = lx - BHALF, hix = lx + BHALF;
    const float loy = ly - BHALF, hiy = ly + BHALF;
    const float loz = lz - BHALF, hiz = lz + BHALF;

    const float* xb = x + (size_t)b * (3 * PP);

    int running = 0;
    for (int base = 0; base < PP; base += 256) {
        const int p = base + tid;
        bool in = false;
        float px = 0.f, py = 0.f, pz = 0.f;
        if (p < PP) {
            px = xb[p];
            py = xb[PP + p];
            pz = xb[2 * PP + p];
            in = (px >= lox) && (px <= hix) &&
                 (py >= loy) && (py <= hiy) &&
                 (pz >= loz) && (pz <= hiz);
        }
        const unsigned mask   = __builtin_amdgcn_ballot_w32(in);
        const int      prefix = __popc(mask & ((1u << lane) - 1u));
        if (lane == 0) s_wcnt[wave] = __popc(mask);
        __syncthreads();

        int woff = 0, ctotal = 0;
#pragma unroll
        for (int w = 0; w < 8; ++w) {
            const int c = s_wcnt[w];
            if (w < wave) woff += c;
            ctotal += c;
        }
        const int rank = running + woff + prefix;
        if (in && rank < NPTS) {
            s_sel[rank]            = px;
            s_sel[NPTS + rank]     = py;
            s_sel[2 * NPTS + rank] = pz;
        }
        running += ctotal;
        __syncthreads();
        if (running >= NPTS) break;   // uniform across the block
    }

    const int count = running < NPTS ? running : NPTS;
    if (count == 0) {                 // empty box -> zero fill
        for (int j = tid; j < 3 * NPTS; j += 256) phi[(size_t)b * 300 + j] = 0.f;
        return;
    }
    if (count < NPTS) {               // wrap (repeat) fill
        for (int j = count + tid; j < NPTS; j += 256) {
            const int src = j % count;
            s_sel[j]            = s_sel[src];
            s_sel[NPTS + j]     = s_sel[NPTS + src];
            s_sel[2 * NPTS + j] = s_sel[2 * NPTS + src];
        }
        __syncthreads();
    }
    for (int j = tid; j < 3 * NPTS; j += 256)
        phi[(size_t)b * 300 + j] = s_sel[j];
}

// ---------------- Phase 2: phi_out = relu(phi @ W1^T + b1) ----------------
// M=B, N=128, K=300. One wave = one 16x16 tile; 8 waves cover N=128.
// A-frag (f32 16x4): lanes 0-15 -> K=k0,k0+1; lanes 16-31 -> K=k0+2,k0+3.
// B-frag (f32 4x16): mirrored; B[k][n] = W1[n][k].
__global__ void gemm_phi_kernel(const float* __restrict__ phi,
                                const float* __restrict__ W1,
                                const float* __restrict__ b1,
                                float* __restrict__ A2)
{
    const int tid  = threadIdx.x;
    const int lane = tid & 31;
    const int wave = tid >> 5;
    const int m0   = blockIdx.x * 16;
    const int n0   = wave * 16;

    const int hi   = lane >> 4;       // 0: K pair 0/1, 1: K pair 2/3
    const int l15  = lane & 15;
    const int arow = m0 + l15;
    const int bcol = n0 + l15;

    const float* aptr = phi + (size_t)arow * 300 + 2 * hi;
    const float* bptr = W1  + (size_t)bcol * 300 + 2 * hi;

    v8f c = {};
    for (int k0 = 0; k0 < 300; k0 += 4) {
        const v2f a = *(const v2f*)(aptr + k0);
        const v2f b = *(const v2f*)(bptr + k0);
        c = wmma_f32_k4(a, b, c);
    }
    const float bias = b1[bcol];
#pragma unroll
    for (int r = 0; r < 8; ++r) {
        const int row = m0 + r + 8 * hi;   // C frag: VGPR r -> M=r / M=r+8
        float v = c[r] + bias;
        A2[(size_t)row * 256 + bcol] = v > 0.f ? v : 0.f;
    }
}

// ---------------- Phase 3: l_out = relu(l @ W2^T + b2), K=3 ----------------
__global__ void lout_kernel(const float* __restrict__ l,
                            const float* __restrict__ W2,
                            const float* __restrict__ b2,
                            float* __restrict__ A2)
{
    const int idx = blockIdx.x * blockDim.x + threadIdx.x;  // B*128 threads
    const int bb  = idx >> 7;
    const int n   = idx & 127;
    const float v = b2[n]
                  + l[bb * 3 + 0] * W2[n * 3 + 0]
                  + l[bb * 3 + 1] * W2[n * 3 + 1]
                  + l[bb * 3 + 2] * W2[n * 3 + 2];
    A2[(size_t)bb * 256 + 128 + n] = v > 0.f ? v : 0.f;
}

// ------- Phase 4: out = relu(A2 @ [W3|W4]^T + b3 + b4), M=B,N=256,K=256 ----
__global__ void gemm_out_kernel(const float* __restrict__ A2,
                                const float* __restrict__ W3,
                                const float* __restrict__ b3,
                                const float* __restrict__ W4,
                                const float* __restrict__ b4,
                                float* __restrict__ out)
{
    const int tid  = threadIdx.x;
    const int lane = tid & 31;
    const int wave = tid >> 5;
    const int m0   = blockIdx.x * 16;
    const int n0   = (blockIdx.y * 8 + wave) * 16;

    const int hi   = lane >> 4;
    const int l15  = lane & 15;
    const int arow = m0 + l15;
    const int bcol = n0 + l15;

    const float* aptr = A2 + (size_t)arow * 256 + 2 * hi;
    const float* w3p  = W3 + (size_t)bcol * 128 + 2 * hi;
    const float* w4p  = W4 + (size_t)bcol * 128 + 2 * hi;

    v8f c = {};
    for (int k0 = 0; k0 < 128; k0 += 4) {          // K 0..127 from W3
        const v2f a = *(const v2f*)(aptr + k0);
        const v2f b = *(const v2f*)(w3p + k0);
        c = wmma_f32_k4(a, b, c);
    }
    for (int k0 = 0; k0 < 128; k0 += 4) {          // K 128..255 from W4
        const v2f a = *(const v2f*)(aptr + 128 + k0);
        const v2f b = *(const v2f*)(w4p + k0);
        c = wmma_f32_k4(a, b, c);
    }
    const float bias = b3[bcol] + b4[bcol];
#pragma unroll
    for (int r = 0; r < 8; ++r) {
        const int row = m0 + r + 8 * hi;
        float v = c[r] + bias;
        out[(size_t)row * 256 + bcol] = v > 0.f ? v : 0.f;
    }
}

// ---------------------------------------------------------------------------
extern "C" void kernel_launch(void* const* d_in, const int* in_sizes, int n_in,
                              void* d_out, int out_size, void* d_ws, size_t ws_size,
                              hipStream_t stream)
{
    const float* x  = (const float*)d_in[0];
    const float* lt = (const float*)d_in[1];
    const float* W1 = (const float*)d_in[2];
    const float* b1 = (const float*)d_in[3];
    const float* W2 = (const float*)d_in[4];
    const float* b2 = (const float*)d_in[5];
    const float* W3 = (const float*)d_in[6];
    const float* b3 = (const float*)d_in[7];
    const float* W4 = (const float*)d_in[8];
    const float* b4 = (const float*)d_in[9];
    float* out = (float*)d_out;

    const int B = in_sizes[1] / 3;     // 4096

    float* phi = (float*)d_ws;              // B*300 floats
    float* A2  = phi + (size_t)B * 300;     // B*256 floats (concat phi_out|l_out)

    foveate_kernel <<<B, 256, 0, stream>>>(x, lt, phi);
    gemm_phi_kernel<<<B / 16, 256, 0, stream>>>(phi, W1, b1, A2);
    lout_kernel    <<<(B * 128) / 256, 256, 0, stream>>>(lt, W2, b2, A2);
    gemm_out_kernel<<<dim3(B / 16, 2), 256, 0, stream>>>(A2, W3, b3, W4, b4, out);
}